// BongardGNN_21466246546228
// MI455X (gfx1250) — compile-verified
//
#include <hip/hip_runtime.h>
#include <hip/hip_bf16.h>

// ---------------------------------------------------------------------------
// BongardGNN on MI455X (gfx1250, wave32)
//
// Pipeline (matches JAX reference):
//   deg   = 1 + scatter_count(dst);  dinv = rsqrt(deg)
//   L1:  hw = x@W1 ; c = b1 + self + scatter(hw*norm) ; x1 = relu(LN(c))+x@Ws1+bs1
//   L2:  hw = x1@W2; c = b2 + self + scatter          ; x2 = relu(LN(c))+x1
//   L3:  out = x1@Ws2+bs2 ; out += b3 + self + scatter(x2@W3 * norm)
//
// GEMMs use V_WMMA_F32_16X16X4_F32 (exact fp32). Scatter uses native
// global_atomic_add_f32. Gathers hit the 192MB L2 (each feature matrix is
// only 51MB), so the irregular phase is the bottleneck, not the GEMMs.
// ---------------------------------------------------------------------------

typedef float v2f __attribute__((ext_vector_type(2)));
typedef float v8f __attribute__((ext_vector_type(8)));

// ---------------------------- degree / norm --------------------------------

__global__ void fill_ones(float* __restrict__ p, int n) {
    int i = blockIdx.x * blockDim.x + threadIdx.x;
    if (i < n) p[i] = 1.0f;  // self-loop contributes 1 to every node's degree
}

__global__ void deg_count(const int* __restrict__ dst, float* __restrict__ deg, int E) {
    int e = blockIdx.x * blockDim.x + threadIdx.x;
    if (e < E) unsafeAtomicAdd(&deg[dst[e]], 1.0f);  // global_atomic_add_f32
}

__global__ void rsqrt_inplace(float* __restrict__ p, int n) {
    int i = blockIdx.x * blockDim.x + threadIdx.x;
    if (i < n) p[i] = rsqrtf(p[i]);
}

// ------------------------------ WMMA GEMM ----------------------------------
// O[N,M] = X[N,K] @ W[K,M] (+ bias). One 16x16 tile per wave.
// f32 16x16x4 operand striping (ISA 7.12.2):
//   A (16x4):  lane<16 -> rows, VGPR{0,1} = K{0,1}; lane>=16 -> K{2,3}
//   B (4x16):  lane<16 -> cols, VGPR{0,1} = K{0,1}; lane>=16 -> K{2,3}
//   C/D vgpr v: row = v + 8*(lane>>4), col = lane&15
template <int K, int M>
__global__ __launch_bounds__(128) void wmma_gemm_f32(
    const float* __restrict__ X, const float* __restrict__ W,
    const float* __restrict__ bias, float* __restrict__ O, int ntiles) {
    const int mtiles = M / 16;
    const int wave = threadIdx.x >> 5;
    const int lane = threadIdx.x & 31;
    const int tile = blockIdx.x * 4 + wave;
    if (tile >= ntiles) return;               // whole-wave exit: EXEC stays all-1s
    const int rt = tile / mtiles;
    const int ct = tile - rt * mtiles;
    const int R = rt * 16, Cb = ct * 16;
    const int half = lane >> 4, idx = lane & 15;

    v8f acc = {0.f, 0.f, 0.f, 0.f, 0.f, 0.f, 0.f, 0.f};
    const float* arow = X + (size_t)(R + idx) * K + 2 * half;
    const float* bcol = W + (size_t)(2 * half) * M + Cb + idx;
#pragma unroll
    for (int k = 0; k < K; k += 4) {
        v2f a, b;
        a.x = arow[k];
        a.y = arow[k + 1];
        b.x = bcol[(size_t)k * M];
        b.y = bcol[(size_t)k * M + M];
        acc = __builtin_amdgcn_wmma_f32_16x16x4_f32(
            /*neg_a=*/false, a, /*neg_b=*/false, b,
            /*c_mod=*/(short)0, acc, /*reuse_a=*/false, /*reuse_b=*/false);
    }

    const float bv = bias ? bias[Cb + idx] : 0.0f;
    float* obase = O + (size_t)(R + 8 * half) * M + Cb + idx;
#pragma unroll
    for (int v = 0; v < 8; ++v) obase[(size_t)v * M] = acc[v] + bv;
}

// ------------------- conv: bias + self-loop (+optional accumulate) ---------
// out[i,f] (=|+)= bias[f] + hw[i,f]*dinv[i]^2
template <bool ACCUM>
__global__ void conv_init(float* __restrict__ out, const float* __restrict__ hw,
                          const float* __restrict__ dinv, const float* __restrict__ bias,
                          int N, int M) {
    int t = blockIdx.x * blockDim.x + threadIdx.x;
    if (t >= N * M) return;
    int i = t / M, f = t - i * M;
    float di = dinv[i];
    float v = bias[f] + hw[t] * di * di;
    out[t] = ACCUM ? out[t] + v : v;
}

// ---------------------- conv: edge scatter (atomic) ------------------------
// out[dst,f] += hw[src,f] * dinv[src]*dinv[dst]; blockDim.x == M
#define EDGES_PER_BLOCK 8
__global__ void conv_scatter(float* __restrict__ out, const float* __restrict__ hw,
                             const float* __restrict__ dinv, const int* __restrict__ src,
                             const int* __restrict__ dst, int E, int M) {
    const int f = threadIdx.x;
    const int e0 = blockIdx.x * EDGES_PER_BLOCK;
#pragma unroll
    for (int k = 0; k < EDGES_PER_BLOCK; ++k) {
        int e = e0 + k;
        if (e >= E) break;
        int s = src[e], d = dst[e];
        if (k + 1 < EDGES_PER_BLOCK && e + 1 < E) {
            // warm L2/L0 for the next gather row (global_prefetch_b8)
            __builtin_prefetch(&hw[(size_t)src[e + 1] * M + f], 0, 1);
        }
        float w = dinv[s] * dinv[d];
        unsafeAtomicAdd(&out[(size_t)d * M + f], hw[(size_t)s * M + f] * w);
    }
}

// ------------------- LayerNorm + ReLU + skip (wave per row) ----------------
// out[row,:] = relu((c-m)*rsqrt(v+eps)*g + be) + skip[row,:]   (H = 128)
__global__ __launch_bounds__(256) void ln_relu_skip(
    const float* __restrict__ c, const float* __restrict__ g,
    const float* __restrict__ be, const float* __restrict__ skip,
    float* __restrict__ out, int N) {
    const int H = 128;
    const int lane = threadIdx.x;  // 0..31 (wave32)
    const int row = blockIdx.x * blockDim.y + threadIdx.y;
    if (row >= N) return;
    const float* crow = c + (size_t)row * H;
    float v0 = crow[lane], v1 = crow[lane + 32], v2 = crow[lane + 64], v3 = crow[lane + 96];
    float s = v0 + v1 + v2 + v3;
    float q = v0 * v0 + v1 * v1 + v2 * v2 + v3 * v3;
#pragma unroll
    for (int m = 16; m > 0; m >>= 1) {
        s += __shfl_xor(s, m, 32);
        q += __shfl_xor(q, m, 32);
    }
    float mean = s * (1.0f / H);
    float var = q * (1.0f / H) - mean * mean;
    float inv = rsqrtf(var + 1e-5f);
    const float* srow = skip + (size_t)row * H;
    float* orow = out + (size_t)row * H;
    float vv[4] = {v0, v1, v2, v3};
#pragma unroll
    for (int j = 0; j < 4; ++j) {
        int f = lane + 32 * j;
        float y = (vv[j] - mean) * inv * g[f] + be[f];
        orow[f] = fmaxf(y, 0.0f) + srow[f];
    }
}

// ------------------------------- launcher ----------------------------------

extern "C" void kernel_launch(void* const* d_in, const int* in_sizes, int n_in,
                              void* d_out, int out_size, void* d_ws, size_t ws_size,
                              hipStream_t stream) {
    const int DIN = 64, H = 128, DOUT = 64;
    const int N = in_sizes[0] / DIN;   // 100000
    const int E = in_sizes[1] / 2;     // 1600000

    const float* x   = (const float*)d_in[0];
    const int*   ei  = (const int*)d_in[1];
    const int*   src = ei;
    const int*   dst = ei + E;
    const float* W1  = (const float*)d_in[2];
    const float* b1  = (const float*)d_in[3];
    const float* W2  = (const float*)d_in[4];
    const float* b2  = (const float*)d_in[5];
    const float* W3  = (const float*)d_in[6];
    const float* b3  = (const float*)d_in[7];
    const float* g1  = (const float*)d_in[8];
    const float* be1 = (const float*)d_in[9];
    const float* g2  = (const float*)d_in[10];
    const float* be2 = (const float*)d_in[11];
    const float* Ws1 = (const float*)d_in[12];
    const float* bs1 = (const float*)d_in[13];
    const float* Ws2 = (const float*)d_in[14];
    const float* bs2 = (const float*)d_in[15];
    float* out = (float*)d_out;

    // workspace layout (floats): dinv[N] | hw[N*H] | c[N*H] | x1[N*H] | x2[N*H]
    float* dinv = (float*)d_ws;
    float* hw   = dinv + N;
    float* cbuf = hw + (size_t)N * H;
    float* x1   = cbuf + (size_t)N * H;
    float* x2   = x1 + (size_t)N * H;

    // ---- degrees -> dinv
    fill_ones<<<(N + 255) / 256, 256, 0, stream>>>(dinv, N);
    deg_count<<<(E + 255) / 256, 256, 0, stream>>>(dst, dinv, E);
    rsqrt_inplace<<<(N + 255) / 256, 256, 0, stream>>>(dinv, N);

    const int ntilesH = (N / 16) * (H / 16);     // 50000
    const int ntilesO = (N / 16) * (DOUT / 16);  // 25000
    const int nmH = N * H;
    const int scatterBlocks = (E + EDGES_PER_BLOCK - 1) / EDGES_PER_BLOCK;
    dim3 lnBlk(32, 8);
    const int lnGrid = (N + 7) / 8;

    // ---- layer 1
    wmma_gemm_f32<64, 128><<<(ntilesH + 3) / 4, 128, 0, stream>>>(x, W1, nullptr, hw, ntilesH);
    conv_init<false><<<(nmH + 255) / 256, 256, 0, stream>>>(cbuf, hw, dinv, b1, N, H);
    conv_scatter<<<scatterBlocks, H, 0, stream>>>(cbuf, hw, dinv, src, dst, E, H);
    wmma_gemm_f32<64, 128><<<(ntilesH + 3) / 4, 128, 0, stream>>>(x, Ws1, bs1, x1, ntilesH);
    ln_relu_skip<<<lnGrid, lnBlk, 0, stream>>>(cbuf, g1, be1, /*skip=*/x1, /*out=*/x1, N);

    // ---- layer 2
    wmma_gemm_f32<128, 128><<<(ntilesH + 3) / 4, 128, 0, stream>>>(x1, W2, nullptr, hw, ntilesH);
    conv_init<false><<<(nmH + 255) / 256, 256, 0, stream>>>(cbuf, hw, dinv, b2, N, H);
    conv_scatter<<<scatterBlocks, H, 0, stream>>>(cbuf, hw, dinv, src, dst, E, H);
    ln_relu_skip<<<lnGrid, lnBlk, 0, stream>>>(cbuf, g2, be2, /*skip=*/x1, /*out=*/x2, N);

    // ---- layer 3: out = x1@Ws2 + bs2 ; out += b3 + self + scatter(x2@W3)
    wmma_gemm_f32<128, 64><<<(ntilesO + 3) / 4, 128, 0, stream>>>(x2, W3, nullptr, hw, ntilesO);
    wmma_gemm_f32<128, 64><<<(ntilesO + 3) / 4, 128, 0, stream>>>(x1, Ws2, bs2, out, ntilesO);
    conv_init<true><<<(N * DOUT + 255) / 256, 256, 0, stream>>>(out, hw, dinv, b3, N, DOUT);
    conv_scatter<<<scatterBlocks, DOUT, 0, stream>>>(out, hw, dinv, src, dst, E, DOUT);
}